// Estimator_2345052144206
// MI455X (gfx1250) — compile-verified
//
#include <hip/hip_runtime.h>

typedef __attribute__((ext_vector_type(2))) float v2f;
typedef __attribute__((ext_vector_type(8))) float v8f;

// ---------------- elementwise helpers ----------------

__global__ void k_fill(float* __restrict__ p, float v, int n) {
    int i = blockIdx.x * blockDim.x + threadIdx.x;
    if (i < n) p[i] = v;
}

__global__ void k_deg(const long long* __restrict__ dst, float* __restrict__ deg, int E) {
    int e = blockIdx.x * blockDim.x + threadIdx.x;
    if (e < E) atomicAdd(&deg[dst[e]], 1.0f);
}

__global__ void k_rsqrt_inplace(float* __restrict__ d, int n) {
    int i = blockIdx.x * blockDim.x + threadIdx.x;
    if (i < n) d[i] = rsqrtf(d[i]);
}

// agg[i,c] = h[i,c] * dinv[i]^2   (self-loop term; also initializes agg)
__global__ void k_self_init(const float* __restrict__ h, const float* __restrict__ dinv,
                            float* __restrict__ agg, long long total) {
    long long idx = (long long)blockIdx.x * blockDim.x + threadIdx.x;
    if (idx >= total) return;
    long long node = idx >> 6;
    float di = dinv[node];
    agg[idx] = h[idx] * (di * di);
}

// one wave per edge; 32 lanes cover 64 feature columns (2 per lane)
__global__ void k_edge_agg(const long long* __restrict__ src, const long long* __restrict__ dst,
                           const float* __restrict__ h, const float* __restrict__ dinv,
                           float* __restrict__ agg, int E) {
    int t = blockIdx.x * blockDim.x + threadIdx.x;
    int e = t >> 5;
    int lane = t & 31;
    if (e >= E) return;
    long long s = src[e];
    long long d = dst[e];
    float w = dinv[s] * dinv[d];
    const float* hr = h + s * 64;
    float* ar = agg + d * 64;
    atomicAdd(&ar[lane],      hr[lane]      * w);
    atomicAdd(&ar[lane + 32], hr[lane + 32] * w);
}

// out[i,c] = relu(agg[i,c] + b[c])
__global__ void k_bias_relu(const float* __restrict__ agg, const float* __restrict__ b,
                            float* __restrict__ out, long long total) {
    long long idx = (long long)blockIdx.x * blockDim.x + threadIdx.x;
    if (idx >= total) return;
    out[idx] = fmaxf(agg[idx] + b[idx & 63], 0.0f);
}

// ---------------- WMMA GEMM: Y[M,64] = A[M,K] @ W[K,64] ----------------
// One wave computes a 16-row x 64-col slab. K in {32, 64}.
// f32 WMMA 16x16x4 layouts (ISA 7.12.2):
//   A(16x4):  element (m,k): lane = (k>>1)*16 + m, vgpr = k&1
//   B(4x16):  element (k,n): lane = (k>>1)*16 + n, vgpr = k&1
//   C/D(16x16): vgpr v: lanes 0-15 -> (M=v,   N=lane),
//                       lanes16-31 -> (M=v+8, N=lane-16)
template <int K>
__global__ void k_gemm_wmma(const float* __restrict__ A, const float* __restrict__ W,
                            float* __restrict__ Y, int M) {
    int wave = blockIdx.x * (blockDim.x >> 5) + (threadIdx.x >> 5);
    int lane = threadIdx.x & 31;
    long long row0 = (long long)wave * 16;
    if (row0 >= M) return;              // wave-uniform: EXEC stays all-ones for WMMA

    int mh = lane & 15;                  // m (for A) or n (for B/C) within half-wave
    int hi = lane >> 4;                  // which half of the wave

    v8f c0 = {}, c1 = {}, c2 = {}, c3 = {};
    const float* arow = A + (row0 + mh) * (long long)K + hi * 2;

#pragma unroll
    for (int kb = 0; kb < K; kb += 4) {
        v2f a;
        a.x = arow[kb];                  // k = kb + hi*2
        a.y = arow[kb + 1];              // k = kb + hi*2 + 1

        const float* wr = W + (size_t)(kb + hi * 2) * 64 + mh;
        v2f b0, b1, b2, b3;
        b0.x = wr[0];      b0.y = wr[64];           // n-tile 0..15
        b1.x = wr[16];     b1.y = wr[64 + 16];      // n-tile 16..31
        b2.x = wr[32];     b2.y = wr[64 + 32];      // n-tile 32..47
        b3.x = wr[48];     b3.y = wr[64 + 48];      // n-tile 48..63

        c0 = __builtin_amdgcn_wmma_f32_16x16x4_f32(false, a, false, b0, (short)0, c0, false, false);
        c1 = __builtin_amdgcn_wmma_f32_16x16x4_f32(false, a, false, b1, (short)0, c1, false, false);
        c2 = __builtin_amdgcn_wmma_f32_16x16x4_f32(false, a, false, b2, (short)0, c2, false, false);
        c3 = __builtin_amdgcn_wmma_f32_16x16x4_f32(false, a, false, b3, (short)0, c3, false, false);
    }

    float* yrow = Y + row0 * 64;
#pragma unroll
    for (int v = 0; v < 8; ++v) {
        long long m = v + hi * 8;
        yrow[m * 64 + mh]      = c0[v];
        yrow[m * 64 + 16 + mh] = c1[v];
        yrow[m * 64 + 32 + mh] = c2[v];
        yrow[m * 64 + 48 + mh] = c3[v];
    }
}

// ---------------- FC head: out[g] = dot(h2[g*896 : +896], Wfc) + bfc ----------------
__global__ void k_fc(const float* __restrict__ h2, const float* __restrict__ Wfc,
                     const float* __restrict__ bfc, float* __restrict__ out, int G) {
    int t = blockIdx.x * blockDim.x + threadIdx.x;
    int g = t >> 5;
    int lane = t & 31;
    if (g >= G) return;
    const float* row = h2 + (size_t)g * 896;
    float acc = 0.0f;
#pragma unroll
    for (int j = 0; j < 28; ++j)
        acc += row[lane + 32 * j] * Wfc[lane + 32 * j];
#pragma unroll
    for (int off = 16; off > 0; off >>= 1)
        acc += __shfl_down(acc, off, 32);
    if (lane == 0) out[g] = acc + bfc[0];
}

// ---------------- launch ----------------

extern "C" void kernel_launch(void* const* d_in, const int* in_sizes, int n_in,
                              void* d_out, int out_size, void* d_ws, size_t ws_size,
                              hipStream_t stream) {
    const float*     x     = (const float*)d_in[0];
    const long long* edges = (const long long*)d_in[1];
    const float*     W1    = (const float*)d_in[2];
    const float*     b1    = (const float*)d_in[3];
    const float*     W2    = (const float*)d_in[4];
    const float*     b2    = (const float*)d_in[5];
    const float*     Wfc   = (const float*)d_in[6];
    const float*     bfc   = (const float*)d_in[7];
    float*           out   = (float*)d_out;

    const int N = in_sizes[0] / 32;   // 700000 nodes
    const int E = in_sizes[1] / 2;    // 4000000 edges
    const long long NH = (long long)N * 64;

    const long long* src = edges;
    const long long* dst = edges + E;

    // workspace: dinv [N] | bufA [N*64] | bufB [N*64]
    char*  ws   = (char*)d_ws;
    float* dinv = (float*)ws;
    size_t off  = ((size_t)N * 4 + 255) & ~(size_t)255;
    float* bufA = (float*)(ws + off);
    float* bufB = (float*)(ws + off + (size_t)NH * 4);

    const int T = 256;
    int nb  = (N + T - 1) / T;
    int ebk = (E + T - 1) / T;
    int elemb = (int)((NH + T - 1) / T);
    long long et = (long long)E * 32;
    int edgewb = (int)((et + T - 1) / T);
    int waves = (N + 15) / 16;
    int gemmb = (waves + 7) / 8;       // 8 waves (256 threads) per block

    // degrees -> dinv
    k_fill<<<nb, T, 0, stream>>>(dinv, 1.0f, N);                 // self-loop
    k_deg<<<ebk, T, 0, stream>>>(dst, dinv, E);
    k_rsqrt_inplace<<<nb, T, 0, stream>>>(dinv, N);

    // ---- layer 1: h1 = relu(gcn(x, W1) + b1) ----
    k_gemm_wmma<32><<<gemmb, 256, 0, stream>>>(x, W1, bufA, N);  // bufA = x@W1
    k_self_init<<<elemb, T, 0, stream>>>(bufA, dinv, bufB, NH);  // bufB = self term
    k_edge_agg<<<edgewb, T, 0, stream>>>(src, dst, bufA, dinv, bufB, E);
    k_bias_relu<<<elemb, T, 0, stream>>>(bufB, b1, bufA, NH);    // h1 -> bufA

    // ---- layer 2: h2 = relu(gcn(h1, W2) + b2) ----
    k_gemm_wmma<64><<<gemmb, 256, 0, stream>>>(bufA, W2, bufB, N); // bufB = h1@W2
    k_self_init<<<elemb, T, 0, stream>>>(bufB, dinv, bufA, NH);
    k_edge_agg<<<edgewb, T, 0, stream>>>(src, dst, bufB, dinv, bufA, E);
    k_bias_relu<<<elemb, T, 0, stream>>>(bufA, b2, bufB, NH);    // h2 -> bufB

    // ---- FC head ----
    int G = out_size;                                            // 50000 graphs
    k_fc<<<((long long)G * 32 + T - 1) / T, T, 0, stream>>>(bufB, Wfc, bfc, out, G);
}